// MultiHeadAttentionCell_49460843380947
// MI455X (gfx1250) — compile-verified
//
#include <hip/hip_runtime.h>
#include <hip/hip_bf16.h>

typedef __attribute__((ext_vector_type(2))) float v2f;
typedef __attribute__((ext_vector_type(8))) float v8f;

#define B_ 2
#define L_ 2048
#define H_ 16
#define C_ 64
#define HC_ (H_ * C_)
#define NEG_INF_ (-1e18f)

// ---------------------------------------------------------------------------
// Kernel 1: scores[b,h,i,j] = sum_c Q[b,i,h,c]*K[b,j,h,c] + edge[h,i,j]
// One wave computes one 16x16 tile via 16x V_WMMA_F32_16X16X4_F32 (K=C=64).
//
// fp32 WMMA register layouts (ISA 7.12.2):
//   A 16x4 : lane l -> row m=l&15, holds K = 2*(l>>4), 2*(l>>4)+1  (float2)
//   B 4x16 : lane l -> col n=l&15, holds K = 2*(l>>4), 2*(l>>4)+1  (float2)
//   C/D    : VGPR r -> row r + 8*(l>>4), col l&15
// ---------------------------------------------------------------------------
__global__ __launch_bounds__(256) void mha_scores_kernel(
    const float* __restrict__ Q, const float* __restrict__ K,
    const float* __restrict__ E, float* __restrict__ S)
{
    const int lane = threadIdx.x;            // 0..31
    const int wave = threadIdx.y;            // 0..7  -> j-tile sub-index
    const int j0   = (blockIdx.x * 8 + wave) * 16;
    const int i0   = blockIdx.y * 16;
    const int bh   = blockIdx.z;
    const int b    = bh / H_;
    const int h    = bh % H_;

    const int m  = lane & 15;
    const int kg = lane >> 4;                // 0 or 1

    const float* qrow = Q + ((size_t)b * L_ + (i0 + m)) * HC_ + h * C_;
    const float* krow = K + ((size_t)b * L_ + (j0 + m)) * HC_ + h * C_;

    v8f acc = {};
#pragma unroll
    for (int c0 = 0; c0 < C_; c0 += 4) {
        v2f a  = *(const v2f*)(qrow + c0 + 2 * kg);   // A[m][c0+2kg .. +1]
        v2f bb = *(const v2f*)(krow + c0 + 2 * kg);   // B[c0+2kg..][n=m]
        acc = __builtin_amdgcn_wmma_f32_16x16x4_f32(
            false, a, false, bb, (short)0, acc, false, false);
    }

    const float* erow = E + ((size_t)h * L_ + i0) * L_ + j0;
    float* srow = S + ((size_t)bh * L_ + i0) * L_ + j0;
#pragma unroll
    for (int r = 0; r < 8; ++r) {
        const int mo = r + 8 * kg;
        srow[(size_t)mo * L_ + m] = acc[r] + erow[(size_t)mo * L_ + m];
    }
}

// ---------------------------------------------------------------------------
// Kernel 2: masked softmax along j.
//   s = mask ? scores/8 : -1e18 ;  w = softmax(s) * mask
// One 256-thread block per (b,h,i) row of 2048 elements.
// ---------------------------------------------------------------------------
__global__ __launch_bounds__(256) void mha_softmax_kernel(
    const float* __restrict__ S, const int* __restrict__ mask,
    float* __restrict__ W)
{
    __shared__ float red[256];
    const int tid = threadIdx.x;
    const int row = blockIdx.x;              // (b*H + h)*L + i
    const int i   = row % L_;
    const int bh  = row / L_;
    const int b   = bh / H_;

    const float* srow = S + (size_t)row * L_;
    const int*   mrow = mask + ((size_t)b * L_ + i) * L_;
    float*       wrow = W + (size_t)row * L_;

    const float invScale = 0.125f;           // 1/sqrt(64)
    float v[L_ / 256];
    int   mk[L_ / 256];

    float lmax = -3.0e38f;
#pragma unroll
    for (int t = 0; t < L_ / 256; ++t) {
        const int j = tid + t * 256;
        mk[t] = mrow[j];
        v[t]  = mk[t] ? srow[j] * invScale : NEG_INF_;
        lmax  = fmaxf(lmax, v[t]);
    }
    red[tid] = lmax;
    __syncthreads();
    for (int off = 128; off > 0; off >>= 1) {
        if (tid < off) red[tid] = fmaxf(red[tid], red[tid + off]);
        __syncthreads();
    }
    const float rmax = red[0];
    __syncthreads();

    float lsum = 0.f;
#pragma unroll
    for (int t = 0; t < L_ / 256; ++t) {
        v[t] = __expf(v[t] - rmax);
        lsum += v[t];
    }
    red[tid] = lsum;
    __syncthreads();
    for (int off = 128; off > 0; off >>= 1) {
        if (tid < off) red[tid] += red[tid + off];
        __syncthreads();
    }
    const float inv = 1.0f / red[0];

#pragma unroll
    for (int t = 0; t < L_ / 256; ++t) {
        const int j = tid + t * 256;
        wrow[j] = mk[t] ? v[t] * inv : 0.0f;
    }
}

// ---------------------------------------------------------------------------
// Kernel 3: context[b,i,h,c] = sum_j attn[b,h,i,j] * V[b,j,h,c]
// One wave computes a full 16x64 output strip (4 c-tiles) so the attn panel
// (the largest read stream, 537 MB) is loaded ONCE and reused by 4
// accumulators: 4 WMMAs share each A load. K-loop over j (2048) in steps of 4.
// Output layout (B,L,H,C) flat == (B,L,H*C) reshape.
// ---------------------------------------------------------------------------
__global__ __launch_bounds__(256) void mha_context_kernel(
    const float* __restrict__ W, const float* __restrict__ V,
    float* __restrict__ O)
{
    const int lane = threadIdx.x;
    const int wave = threadIdx.y;            // 0..7 -> i-tile sub-index
    const int it = blockIdx.x * 8 + wave;
    const int bh = blockIdx.y;
    const int b  = bh / H_;
    const int h  = bh % H_;
    const int i0 = it * 16;

    const int m  = lane & 15;
    const int kg = lane >> 4;

    const float* arow  = W + ((size_t)bh * L_ + (i0 + m)) * L_;
    const float* vbase = V + (size_t)b * L_ * HC_ + h * C_ + m;   // + j*HC_ + 16*ct

    v8f acc0 = {}, acc1 = {}, acc2 = {}, acc3 = {};
#pragma unroll 2
    for (int j0 = 0; j0 < L_; j0 += 4) {
        v2f a = *(const v2f*)(arow + j0 + 2 * kg);        // A[m][j0+2kg..+1]
        const float* vp0 = vbase + (size_t)(j0 + 2 * kg) * HC_;   // row k0
        const float* vp1 = vp0 + HC_;                             // row k0+1
        v2f b0 = { vp0[0],  vp1[0]  };                    // B[k][n=m]
        v2f b1 = { vp0[16], vp1[16] };                    // B[k][n=m+16]
        v2f b2 = { vp0[32], vp1[32] };                    // B[k][n=m+32]
        v2f b3 = { vp0[48], vp1[48] };                    // B[k][n=m+48]
        acc0 = __builtin_amdgcn_wmma_f32_16x16x4_f32(false, a, false, b0, (short)0, acc0, false, false);
        acc1 = __builtin_amdgcn_wmma_f32_16x16x4_f32(false, a, false, b1, (short)0, acc1, false, false);
        acc2 = __builtin_amdgcn_wmma_f32_16x16x4_f32(false, a, false, b2, (short)0, acc2, false, false);
        acc3 = __builtin_amdgcn_wmma_f32_16x16x4_f32(false, a, false, b3, (short)0, acc3, false, false);
    }

    float* orow = O + ((size_t)b * L_ + i0) * HC_ + h * C_;
#pragma unroll
    for (int r = 0; r < 8; ++r) {
        const int mo = r + 8 * kg;
        float* p = orow + (size_t)mo * HC_ + m;
        p[0]  = acc0[r];
        p[16] = acc1[r];
        p[32] = acc2[r];
        p[48] = acc3[r];
    }
}

extern "C" void kernel_launch(void* const* d_in, const int* in_sizes, int n_in,
                              void* d_out, int out_size, void* d_ws, size_t ws_size,
                              hipStream_t stream) {
    const float* Q    = (const float*)d_in[0];
    const float* K    = (const float*)d_in[1];
    const float* Vv   = (const float*)d_in[2];
    const int*   mask = (const int*)d_in[3];
    const float* E    = (const float*)d_in[4];

    float* out = (float*)d_out;
    float* ctx = out;                                        // B*L*H*C
    float* S   = out + (size_t)B_ * L_ * HC_;                // B*H*L*L
    float* W   = S   + (size_t)B_ * H_ * L_ * L_;            // B*H*L*L

    dim3 blk(32, 8);
    // scores + edge: grid (j-tiles/8, i-tiles, B*H)
    mha_scores_kernel<<<dim3(L_ / 16 / 8, L_ / 16, B_ * H_), blk, 0, stream>>>(Q, K, E, S);
    // masked softmax: one block per row
    mha_softmax_kernel<<<dim3(B_ * H_ * L_), dim3(256), 0, stream>>>(S, mask, W);
    // attn @ V: grid (i-tiles/8, B*H); each wave -> one 16x64 output strip
    mha_context_kernel<<<dim3(L_ / 16 / 8, B_ * H_), blk, 0, stream>>>(W, Vv, ctx);
}